// OptimizedSeriesDecomp_8512625181022
// MI455X (gfx1250) — compile-verified
//
#include <hip/hip_runtime.h>
#include <hip/hip_bf16.h>
#include <stdint.h>

// Series decomposition, B=32, L=4096, C=512, kernels {7,25,49}, replicate pad.
// Streaming-bandwidth bound: 768 MB min traffic -> ~33us @ 23.3 TB/s.
// CDNA5 path: async global->LDS staging (ASYNCcnt) + sliding sums + NT stores.

#define BATCH 32
#define LEN   4096
#define CHAN  512
#define TL    64          // L-tile per block
#define HALO  24          // (49-1)/2
#define ROWS  (TL + 2 * HALO)   // 112
#define CPB   128         // channels per block
// LDS: ROWS * CPB * 4 = 57344 bytes

__global__ __launch_bounds__(CPB) void decomp_kernel(
    const float* __restrict__ x,
    float* __restrict__ res,
    float* __restrict__ trend) {
  __shared__ float sm[ROWS * CPB];

  const int t    = threadIdx.x;       // channel within group, 0..127
  const int tile = blockIdx.x;        // 0..63  (L tiles)
  const int cg   = blockIdx.y;        // 0..3   (channel groups)
  const int b    = blockIdx.z;        // 0..31  (batch)
  const int l0   = tile * TL;
  const int c0   = cg * CPB;
  const size_t baseB = (size_t)b * LEN * CHAN;

  // ---- Stage tile into LDS with async tensor-path loads (B128 chunks) ----
  // ROWS rows x (CPB*4 bytes) = 112 x 512B; 16B chunks: 112*32 = 3584 chunks,
  // 28 per thread. Replicate padding handled by clamping the source row.
  #pragma unroll
  for (int j = 0; j < (ROWS * (CPB / 4)) / CPB; ++j) {   // 28 iterations
    const int idx = t + j * CPB;
    const int row = idx >> 5;          // 0..111
    const int ck  = idx & 31;          // 16B chunk within the row
    int l = l0 - HALO + row;
    l = l < 0 ? 0 : (l > (LEN - 1) ? (LEN - 1) : l);
    const float* gp = x + baseB + (size_t)l * CHAN + (size_t)(c0 + ck * 4);
    const unsigned ldsa =
        (unsigned)(uintptr_t)(&sm[row * CPB + ck * 4]);   // low 32b of generic = LDS offset
    const unsigned long long ga = (unsigned long long)(uintptr_t)gp;
    asm volatile("global_load_async_to_lds_b128 %0, %1, off"
                 :: "v"(ldsa), "v"(ga)
                 : "memory");
  }
  asm volatile("s_wait_asynccnt 0" ::: "memory");
  __syncthreads();

  // ---- Sliding-window sums along L for this thread's channel column ----
  float s7 = 0.f, s25 = 0.f, s49 = 0.f;
  #pragma unroll
  for (int r = 0; r < 49; ++r) {
    const float v = sm[r * CPB + t];
    s49 += v;
    if (r >= 12 && r <= 36) s25 += v;   // compile-time predicates (unrolled)
    if (r >= 21 && r <= 27) s7  += v;
  }

  const float C7  = 1.0f / 7.0f;
  const float C25 = 1.0f / 25.0f;
  const float C49 = 1.0f / 49.0f;
  const float C3  = 1.0f / 3.0f;

  size_t outIdx = baseB + (size_t)l0 * CHAN + (size_t)(c0 + t);
  #pragma unroll 4
  for (int i = 0; i < TL; ++i) {
    const float xc = sm[(i + HALO) * CPB + t];
    const float tr = (s7 * C7 + s25 * C25 + s49 * C49) * C3;
    __builtin_nontemporal_store(xc - tr, res + outIdx);
    __builtin_nontemporal_store(tr, trend + outIdx);
    outIdx += CHAN;
    // advance windows to center i+1
    s49 += sm[(i + 49) * CPB + t] - sm[(i +  0) * CPB + t];
    s25 += sm[(i + 37) * CPB + t] - sm[(i + 12) * CPB + t];
    s7  += sm[(i + 28) * CPB + t] - sm[(i + 21) * CPB + t];
  }
}

extern "C" void kernel_launch(void* const* d_in, const int* in_sizes, int n_in,
                              void* d_out, int out_size, void* d_ws, size_t ws_size,
                              hipStream_t stream) {
  (void)in_sizes; (void)n_in; (void)out_size; (void)d_ws; (void)ws_size;
  const float* x = (const float*)d_in[0];
  float* res   = (float*)d_out;
  float* trend = res + (size_t)BATCH * LEN * CHAN;

  dim3 grid(LEN / TL, CHAN / CPB, BATCH);   // 64 x 4 x 32
  dim3 block(CPB);                          // 128 threads = 4 waves
  decomp_kernel<<<grid, block, 0, stream>>>(x, res, trend);
}